// GHMC_34995393527862
// MI455X (gfx1250) — compile-verified
//
#include <hip/hip_runtime.h>
#include <hip/hip_bf16.h>
#include <stdint.h>

#define BINS 10
#define NROWS 262144
#define NCOLS 128
#define BLOCKS 4096
#define THREADS 256
#define NPART  (BLOCKS)   // one partial per block

typedef float v2f __attribute__((ext_vector_type(2)));
typedef float v8f __attribute__((ext_vector_type(8)));

// ---------------------------------------------------------------------------
// Per-element math helpers (lean: 3 transcendentals + ~10 VALU per element)
// ---------------------------------------------------------------------------
__device__ __forceinline__ float fast_exp_neg(float a) {
    // a >= 0 ; exp(-a) via v_exp_f32 (exp2)
    return __expf(-a);
}

// ---------------------------------------------------------------------------
// Kernel 0: zero the 10 histogram counters in workspace (ws is poisoned 0xAA)
// ---------------------------------------------------------------------------
__global__ void ghmc_init(unsigned* __restrict__ counts) {
    if (threadIdx.x < BINS) counts[threadIdx.x] = 0u;
}

// ---------------------------------------------------------------------------
// Kernel 1: histogram pass.
// Wave-per-row: lane L handles columns 4L..4L+3 (float4, 16B aligned).
// Packed 64-bit bin counters: 10 bins x 6 bits; flush every 15 rows
// (15 rows * 4 elems = 60 < 63, no 6-bit overflow possible).
// ---------------------------------------------------------------------------
__global__ void ghmc_hist(const float* __restrict__ in,
                          const long long* __restrict__ tgt,
                          const float* __restrict__ mask,
                          unsigned* __restrict__ counts) {
    const int tid    = blockIdx.x * blockDim.x + threadIdx.x;
    const int lane   = threadIdx.x & 31;
    const int wave   = tid >> 5;
    const int nwaves = (gridDim.x * blockDim.x) >> 5;

    unsigned loc[BINS];
#pragma unroll
    for (int b = 0; b < BINS; ++b) loc[b] = 0u;

    unsigned long long pack = 0ull;
    int budget = 15;

    for (int row = wave; row < NROWS; row += nwaves) {
        const float mv = mask[row];                 // wave-uniform load
        const float4 x = *(const float4*)(in + (size_t)row * NCOLS + lane * 4);
        if (mv > 0.0f) {
            const int   t  = (int)tgt[row];         // wave-uniform load
            const int   c0 = lane * 4;
            float xv[4] = {x.x, x.y, x.z, x.w};
#pragma unroll
            for (int j = 0; j < 4; ++j) {
                const float xi = xv[j];
                const float tv = (c0 + j == t) ? 1.0f : 0.0f;
                const float a  = fabsf(xi);
                const float e  = fast_exp_neg(a);               // v_exp
                const float r  = __builtin_amdgcn_rcpf(1.0f + e); // v_rcp
                const float sg = (xi >= 0.0f) ? r : e * r;      // sigmoid(x)
                const float g  = fabsf(sg - tv);
                int b = (int)(g * 10.0f);                       // trunc == floor (g>=0)
                b = (b > BINS - 1) ? (BINS - 1) : b;
                pack += 1ull << (6 * b);
            }
            if (--budget == 0) {
#pragma unroll
                for (int b = 0; b < BINS; ++b)
                    loc[b] += (unsigned)((pack >> (6 * b)) & 63ull);
                pack = 0ull;
                budget = 15;
            }
        }
    }
#pragma unroll
    for (int b = 0; b < BINS; ++b)
        loc[b] += (unsigned)((pack >> (6 * b)) & 63ull);

    // Block-level reduction in LDS, then 10 global u32 atomics per block.
    __shared__ unsigned sh[BINS];
    if (threadIdx.x < BINS) sh[threadIdx.x] = 0u;
    __syncthreads();
#pragma unroll
    for (int b = 0; b < BINS; ++b)
        if (loc[b]) atomicAdd(&sh[b], loc[b]);
    __syncthreads();
    if (threadIdx.x < BINS) atomicAdd(&counts[threadIdx.x], sh[threadIdx.x]);
}

// ---------------------------------------------------------------------------
// Kernel 2: per-bin coefficients. coef[b] folds w_per_bin, /n and /tot:
//   tot = max(sum counts, 1); n = #nonempty bins
//   coef[b] = counts[b]>0 ? (tot/max(counts[b],1)) * (1/n) * (1/tot) : 0
// ---------------------------------------------------------------------------
__global__ void ghmc_coef(const unsigned* __restrict__ counts,
                          float* __restrict__ coef) {
    if (threadIdx.x == 0 && blockIdx.x == 0) {
        float c[BINS];
        float tot = 0.0f;
        int   n   = 0;
#pragma unroll
        for (int b = 0; b < BINS; ++b) {
            c[b] = (float)counts[b];
            tot += c[b];
            if (counts[b] > 0u) ++n;
        }
        tot = fmaxf(tot, 1.0f);
        const float invn   = (n > 0) ? (1.0f / (float)n) : 1.0f;
        const float invtot = 1.0f / tot;
#pragma unroll
        for (int b = 0; b < BINS; ++b)
            coef[b] = (counts[b] > 0u)
                        ? (tot / fmaxf(c[b], 1.0f)) * invn * invtot
                        : 0.0f;
    }
}

// ---------------------------------------------------------------------------
// Kernel 3: weighted stable-BCE sum. Pass 2 re-reads the same 134 MB which
// fits in the 192 MB L2 (RT temporal hint), so it should be L2-resident.
// log1p(exp(-|x|)) = -log(r) reuses r from the sigmoid.
// One deterministic partial per block.
// ---------------------------------------------------------------------------
__global__ void ghmc_loss(const float* __restrict__ in,
                          const long long* __restrict__ tgt,
                          const float* __restrict__ mask,
                          const float* __restrict__ coef,
                          float* __restrict__ partials) {
    __shared__ float scoef[BINS];
    if (threadIdx.x < BINS) scoef[threadIdx.x] = coef[threadIdx.x];
    __syncthreads();

    const int tid    = blockIdx.x * blockDim.x + threadIdx.x;
    const int lane   = threadIdx.x & 31;
    const int wave   = tid >> 5;
    const int nwaves = (gridDim.x * blockDim.x) >> 5;

    float acc = 0.0f;
    for (int row = wave; row < NROWS; row += nwaves) {
        const float mv = mask[row];
        const float4 x = *(const float4*)(in + (size_t)row * NCOLS + lane * 4);
        if (mv > 0.0f) {
            const int t  = (int)tgt[row];
            const int c0 = lane * 4;
            float xv[4] = {x.x, x.y, x.z, x.w};
#pragma unroll
            for (int j = 0; j < 4; ++j) {
                const float xi = xv[j];
                const float tv = (c0 + j == t) ? 1.0f : 0.0f;
                const float a  = fabsf(xi);
                const float e  = fast_exp_neg(a);
                const float r  = __builtin_amdgcn_rcpf(1.0f + e);
                const float sg = (xi >= 0.0f) ? r : e * r;
                const float g  = fabsf(sg - tv);
                int b = (int)(g * 10.0f);
                b = (b > BINS - 1) ? (BINS - 1) : b;
                const float bce = fmaxf(xi, 0.0f) - xi * tv - __logf(r);
                acc += scoef[b] * bce;                   // ds_load broadcast
            }
        }
    }

    // wave reduce (wave32), then 8-wave LDS reduce -> one partial per block
#pragma unroll
    for (int off = 16; off > 0; off >>= 1)
        acc += __shfl_xor(acc, off, 32);

    __shared__ float red[THREADS / 32];
    if (lane == 0) red[threadIdx.x >> 5] = acc;
    __syncthreads();
    if (threadIdx.x == 0) {
        float s = 0.0f;
#pragma unroll
        for (int w = 0; w < THREADS / 32; ++w) s += red[w];
        partials[blockIdx.x] = s;
    }
}

// ---------------------------------------------------------------------------
// Kernel 4: final reduction of NPART partials via V_WMMA_F32_16X16X4_F32.
// A (16x4 f32, 2 VGPRs/lane) carries 64 partials per iteration — the
// (lane, vgpr) -> (M,K) mapping is a bijection onto 64 A-slots, and any
// permutation is fine under summation. B = ones, so D[m][n] = rowsum_m,
// accumulated in the f32 C across all iterations. Then each lane sums its
// 8 D VGPRs (= rowsums 0..7 for lanes<16, 8..15 for lanes>=16) and one
// shfl_xor(16) completes the 16x16 tile sum. Exact f32 math, EXEC all-1s.
// ---------------------------------------------------------------------------
__global__ void ghmc_final(const float* __restrict__ partials,
                           float* __restrict__ out) {
    const int lane = threadIdx.x;   // launched with exactly 32 threads
    v8f c = {};                     // f32 accumulator (16x16)
    v2f ones;
    ones.x = 1.0f; ones.y = 1.0f;

    for (int base = 0; base < NPART; base += 64) {
        v2f a;
        a.x = partials[base + lane * 2 + 0];
        a.y = partials[base + lane * 2 + 1];
        // D = A x ones + C  (8 args: neg_a, A, neg_b, B, c_mod, C, reuse_a, reuse_b)
        c = __builtin_amdgcn_wmma_f32_16x16x4_f32(
                false, a, false, ones, (short)0, c, false, false);
    }

    float s = c[0] + c[1] + c[2] + c[3] + c[4] + c[5] + c[6] + c[7];
    s += __shfl_xor(s, 16, 32);
    if (lane == 0) *out = s;        // coef already folded /n and /tot
}

// ---------------------------------------------------------------------------
// Launch
// ---------------------------------------------------------------------------
extern "C" void kernel_launch(void* const* d_in, const int* in_sizes, int n_in,
                              void* d_out, int out_size, void* d_ws, size_t ws_size,
                              hipStream_t stream) {
    (void)in_sizes; (void)n_in; (void)out_size; (void)ws_size;

    const float*     input  = (const float*)d_in[0];
    const long long* target = (const long long*)d_in[1];   // int64 per reference
    const float*     maskp  = (const float*)d_in[2];
    float*           outp   = (float*)d_out;

    uint8_t*  ws       = (uint8_t*)d_ws;
    unsigned* counts   = (unsigned*)(ws);          // 10 x u32
    float*    coef     = (float*)(ws + 64);        // 10 x f32
    float*    partials = (float*)(ws + 128);       // NPART x f32

    ghmc_init <<<1, 32, 0, stream>>>(counts);
    ghmc_hist <<<BLOCKS, THREADS, 0, stream>>>(input, target, maskp, counts);
    ghmc_coef <<<1, 32, 0, stream>>>(counts, coef);
    ghmc_loss <<<BLOCKS, THREADS, 0, stream>>>(input, target, maskp, coef, partials);
    ghmc_final<<<1, 32, 0, stream>>>(partials, outp);
}